// EquivariantUpsampleConv_51926154609217
// MI455X (gfx1250) — compile-verified
//
#include <hip/hip_runtime.h>

typedef float v2f __attribute__((ext_vector_type(2)));
typedef float v8f __attribute__((ext_vector_type(8)));

#define NPIX (1024*1024)
#define NINE9 (NPIX*9)
static const double PI_D = 3.14159265358979323846;

// ======================= setup-side math (device doubles) =======================
__device__ __forceinline__ int imax3(int a,int b,int c){ int m=a>b?a:b; return m>c?m:c; }
__device__ __forceinline__ int imin3(int a,int b,int c){ int m=a<b?a:b; return m<c?m:c; }

__device__ __forceinline__ double factd(int n){ double r=1.0; for(int i=2;i<=n;++i) r*=(double)i; return r; }

__device__ double cgc(int j1,int m1,int j2,int m2,int j3,int m3){
  if (m1+m2 != m3) return 0.0;
  double pre = (2.0*j3+1.0)*factd(j1+j2-j3)*factd(j1-j2+j3)*factd(-j1+j2+j3)/factd(j1+j2+j3+1);
  pre *= factd(j1+m1)*factd(j1-m1)*factd(j2+m2)*factd(j2-m2)*factd(j3+m3)*factd(j3-m3);
  int kmin = imax3(0, j2-j3-m1, j1-j3+m2);
  int kmax = imin3(j1+j2-j3, j1-m1, j2+m2);
  double s = 0.0;
  for(int k=kmin;k<=kmax;++k){
    double d = factd(k)*factd(j1+j2-j3-k)*factd(j1-m1-k)*factd(j2+m2-k)*factd(j3-j2+m1+k)*factd(j3-j1-m2+k);
    s += ((k&1)? -1.0 : 1.0)/d;
  }
  return sqrt(pre)*s;
}

struct cplx { double re, im; };
__device__ __forceinline__ cplx mkc(double r,double i){ cplx c; c.re=r; c.im=i; return c; }
__device__ __forceinline__ cplx cmul(cplx a, cplx b){ return mkc(a.re*b.re - a.im*b.im, a.re*b.im + a.im*b.re); }

// Nonzero entries of row r of complex->real SH change-of-basis q(l) (incl. (-i)^l phase).
__device__ int qrow(int l, int r, int* cols, cplx* vals){
  cplx ph;
  switch(l & 3){
    case 0:  ph = mkc( 1.0, 0.0); break;
    case 1:  ph = mkc( 0.0,-1.0); break;
    case 2:  ph = mkc(-1.0, 0.0); break;
    default: ph = mkc( 0.0, 1.0); break;
  }
  const double s2 = 0.70710678118654752440;
  int m = r - l;
  if (m == 0){ cols[0]=l; vals[0]=ph; return 1; }
  if (m > 0){
    double sg = (m&1)? -1.0 : 1.0;
    cols[0]=l+m; vals[0]=cmul(ph, mkc(sg*s2,0.0));
    cols[1]=l-m; vals[1]=cmul(ph, mkc(s2,0.0));
    return 2;
  }
  int mm=-m; double sg=(mm&1)? -1.0 : 1.0;
  cols[0]=l-mm; vals[0]=cmul(ph, mkc(0.0,s2));
  cols[1]=l+mm; vals[1]=cmul(ph, mkc(0.0,-sg*s2));
  return 2;
}

__device__ double w3jv(int l1,int l2,int l3,int i,int j,int k){
  int m1=i-l1, m2=j-l2, m3=k-l3;
  if (m1+m2+m3 != 0) return 0.0;
  double sg = ((l1-l2-m3)&1)? -1.0 : 1.0;
  return sg / sqrt((double)(2*l3+1)) * cgc(l1,m1,l2,m2,l3,-m3);
}

// Real-basis Wigner-3j coefficient CG[l1,l2,l3][a,b,c]
__device__ double cgreal(int l1,int l2,int l3,int a,int b,int c){
  int c1[2],c2[2],c3[2]; cplx v1[2],v2[2],v3[2];
  int n1=qrow(l1,a,c1,v1), n2=qrow(l2,b,c2,v2), n3=qrow(l3,c,c3,v3);
  double acc=0.0;
  for(int x=0;x<n1;++x) for(int y=0;y<n2;++y) for(int z=0;z<n3;++z){
    double w = w3jv(l1,l2,l3,c1[x],c2[y],c3[z]);
    if (w != 0.0){
      cplx t = cmul(cmul(v1[x],v2[y]),v3[z]);
      acc += t.re * w;
    }
  }
  return acc;
}

// SH_KERNEL[k][col]; z==0 for all 4 dirs -> cols 2,4,5 vanish, x^2==y^2 -> col5 vanishes.
__device__ double shv(int k, int col){
  const double s = 0.70710678118654752440;
  double x = (k&1)? s : -s;
  double y = (k&2)? s : -s;
  switch(col){
    case 0: return 0.5/sqrt(PI_D);
    case 1: return 0.5*sqrt(15.0/PI_D)*x*y;
    case 3: return -0.25*sqrt(5.0/PI_D);
    default: return 0.0;
  }
}

// Combined aggregation matrix A[K=(k*22+a)][c], 88x22
__device__ float Aval(int K, int c, const float* w_agg){
  const int k = K/22, a = K%22;
  const int L1[6]={4,4,4,6,6,6}, L2[6]={0,2,2,0,2,2}, L3[6]={4,4,6,6,4,6};
  double acc = 0.0;
  for(int i=0;i<6;++i){
    int l1=L1[i], l2=L2[i], l3=L3[i];
    int a0=(l1==4)?0:9, ad=(l1==4)?9:13;
    int b0=(l2==0)?0:1, bd=(l2==0)?1:5;
    int c0=(l3==4)?0:9, cd=(l3==4)?9:13;
    if (a<a0 || a>=a0+ad || c<c0 || c>=c0+cd) continue;
    double alpha = sqrt((2.0*l3+1.0)/3.0);
    double s = 0.0;
    for(int b=0;b<bd;++b) s += shv(k, b0+b) * cgreal(l1,l2,l3, a-a0, b, c-c0);
    acc += (double)w_agg[i] * alpha * s;
  }
  return (float)acc;
}

// Combined tensor-product matrix T2[K=(a*22+b)][c], 484x22
__device__ float Tval(int K, int c, const float* w_tp){
  const int a = K/22, b = K%22;
  const int L1[8]={4,4,4,4,6,6,6,6}, L2[8]={4,4,6,6,4,4,6,6}, L3[8]={4,6,4,6,4,6,4,6};
  double acc = 0.0;
  for(int i=0;i<8;++i){
    int l1=L1[i], l2=L2[i], l3=L3[i];
    int a0=(l1==4)?0:9, ad=(l1==4)?9:13;
    int b0=(l2==4)?0:9, bd=(l2==4)?9:13;
    int c0=(l3==4)?0:9, cd=(l3==4)?9:13;
    if (a<a0 || a>=a0+ad || b<b0 || b>=b0+bd || c<c0 || c>=c0+cd) continue;
    double alpha = sqrt((2.0*l3+1.0)/4.0);
    acc += (double)w_tp[i] * alpha * cgreal(l1,l2,l3, a-a0, b-b0, c-c0);
  }
  return (float)acc;
}

// Build both weight matrices directly in V_WMMA_F32_16X16X4_F32 B-fragment layout:
//   fragment (kc, ntile, lane): VGPR j holds element (K = 4*kc + (lane<16 ? j : 2+j),
//                                                     N = ntile*16 + (lane&15))
// ws layout (floats): [0, 2816)  = A-frags  (22 kc x 2 nt x 32 lanes x 2)
//                     [2816, ..) = T2-frags (121 kc x 2 nt x 32 lanes x 2)
__global__ void eq_setup(const float* __restrict__ w_agg, const float* __restrict__ w_tp,
                         float* __restrict__ frg){
  int id = blockIdx.x*blockDim.x + threadIdx.x;
  const int NA = 22*2*32;    // 1408 float2 slots
  const int NT = 121*2*32;   // 7744 float2 slots
  if (id < NA){
    int kc = id >> 6, nt = (id >> 5) & 1, lane = id & 31;
    #pragma unroll
    for(int j=0;j<2;++j){
      int K  = 4*kc + ((lane<16)? j : 2+j);
      int Nc = nt*16 + (lane & 15);
      frg[id*2+j] = (Nc < 22) ? Aval(K, Nc, w_agg) : 0.f;
    }
  } else if (id < NA+NT){
    int t = id - NA;
    int kc = t >> 6, nt = (t >> 5) & 1, lane = t & 31;
    #pragma unroll
    for(int j=0;j<2;++j){
      int K  = 4*kc + ((lane<16)? j : 2+j);
      int Nc = nt*16 + (lane & 15);
      frg[NA*2 + t*2 + j] = (Nc < 22) ? Tval(K, Nc, w_tp) : 0.f;
    }
  }
}

__global__ void eq_tail(float* __restrict__ p){ p[0] = 1024.0f; p[1] = 1024.0f; }

// ======================= main kernel =======================
// Per wave: 16 consecutive output pixels (WMMA M dim). Stage A: 22 chained
// f32 WMMAs (K=88) x 2 N-tiles; context -> LDS; stage B: 121 chained WMMAs
// (K=484, outer-product A fragments built from LDS) x 2 N-tiles; residual add.
__global__ __launch_bounds__(256) void eq_main(const float* __restrict__ f4,
                                               const float* __restrict__ f6,
                                               const float* __restrict__ frg,
                                               float* __restrict__ out){
  __shared__ float pf[8][16][44];           // [wave][pixel][feat(0..21) | ctx(22..43)]
  const int lane = threadIdx.x & 31;
  const int wave = threadIdx.x >> 5;
  const int g    = blockIdx.x * 8 + wave;   // 65536 groups of 16 pixels
  const int p    = lane & 15;               // this lane's pixel (M index)
  const bool lo  = lane < 16;

  const int n  = g*16 + p;
  const int y  = n >> 10;
  const int x  = n & 1023;
  const int y1 = (y < 1023) ? y+1 : 1023;   // edge-pad clamp
  const int x1 = (x < 1023) ? x+1 : 1023;
  const int by = y>>2, bx = x>>2, by1 = y1>>2, bx1 = x1>>2;   // /R base coords
  int bp[4];
  bp[0] = by*256+bx; bp[1] = by*256+bx1; bp[2] = by1*256+bx; bp[3] = by1*256+bx1;
  int o4[4], o6[4];
  #pragma unroll
  for(int k=0;k<4;++k){ o4[k]=bp[k]*9; o6[k]=bp[k]*13; }

  // stage pixel-p feature vector into LDS (feat_hr[n] == feat at bp[0])
  if (lo){
    #pragma unroll
    for(int c=0;c<11;++c) pf[wave][p][c] = (c<9)? f4[o4[0]+c] : f6[o6[0]+c-9];
  } else {
    #pragma unroll
    for(int c=11;c<22;++c) pf[wave][p][c] = f6[o6[0]+c-9];
  }

  // ---------------- stage A: context = patches(Nx88) @ A(88x22) ----------------
  v8f ca = {0.f,0.f,0.f,0.f,0.f,0.f,0.f,0.f};
  v8f cb = {0.f,0.f,0.f,0.f,0.f,0.f,0.f,0.f};
  #pragma unroll
  for(int kc=0;kc<22;++kc){
    const int qa=4*kc, qb=4*kc+1, qc=4*kc+2, qd=4*kc+3;
    // K index q = k*22 + a : patch value = feat[bp[k]][a] (split across f4/f6)
    const float* pa = ((qa%22)<9) ? (f4 + o4[qa/22] + (qa%22)) : (f6 + o6[qa/22] + (qa%22)-9);
    const float* pb = ((qb%22)<9) ? (f4 + o4[qb/22] + (qb%22)) : (f6 + o6[qb/22] + (qb%22)-9);
    const float* pc = ((qc%22)<9) ? (f4 + o4[qc/22] + (qc%22)) : (f6 + o6[qc/22] + (qc%22)-9);
    const float* pd = ((qd%22)<9) ? (f4 + o4[qd/22] + (qd%22)) : (f6 + o6[qd/22] + (qd%22)-9);
    v2f afr;
    afr.x = *(lo ? pa : pc);                 // A-frag VGPR0: K=4kc | K=4kc+2
    afr.y = *(lo ? pb : pd);                 // A-frag VGPR1: K=4kc+1 | K=4kc+3
    v2f b0 = *(const v2f*)(frg + ((kc*2+0)*32 + lane)*2);
    v2f b1 = *(const v2f*)(frg + ((kc*2+1)*32 + lane)*2);
    ca = __builtin_amdgcn_wmma_f32_16x16x4_f32(false, afr, false, b0, (short)0, ca, false, false);
    cb = __builtin_amdgcn_wmma_f32_16x16x4_f32(false, afr, false, b1, (short)0, cb, false, false);
  }

  // scatter context (C layout: lanes<16 -> (M=r, N=lane); lanes>=16 -> (M=8+r, N=lane-16))
  {
    const int nn = lane & 15;
    #pragma unroll
    for(int r=0;r<8;++r){
      int M = lo ? r : 8+r;
      pf[wave][M][22+nn] = ca[r];
      if (nn < 6) pf[wave][M][22+16+nn] = cb[r];
    }
  }
  __syncthreads();

  // ---------------- stage B: out = P(Nx484) @ T2(484x22), P = feat (x) ctx ----------------
  v8f oa = {0.f,0.f,0.f,0.f,0.f,0.f,0.f,0.f};
  v8f ob = {0.f,0.f,0.f,0.f,0.f,0.f,0.f,0.f};
  const float* tb  = frg + 2816;
  const float* pfp = &pf[wave][p][0];
  #pragma unroll
  for(int kc=0;kc<121;++kc){
    const int qa=4*kc, qb=4*kc+1, qc=4*kc+2, qd=4*kc+3;
    // P[q] = feat[q/22] * ctx[q%22], both from this pixel's LDS row
    float fe0 = *(lo ? (pfp + qa/22)        : (pfp + qc/22));
    float ce0 = *(lo ? (pfp + 22 + (qa%22)) : (pfp + 22 + (qc%22)));
    float fe1 = *(lo ? (pfp + qb/22)        : (pfp + qd/22));
    float ce1 = *(lo ? (pfp + 22 + (qb%22)) : (pfp + 22 + (qd%22)));
    v2f afr; afr.x = fe0*ce0; afr.y = fe1*ce1;
    v2f b0 = *(const v2f*)(tb + ((kc*2+0)*32 + lane)*2);
    v2f b1 = *(const v2f*)(tb + ((kc*2+1)*32 + lane)*2);
    oa = __builtin_amdgcn_wmma_f32_16x16x4_f32(false, afr, false, b0, (short)0, oa, false, false);
    ob = __builtin_amdgcn_wmma_f32_16x16x4_f32(false, afr, false, b1, (short)0, ob, false, false);
  }

  // ---------------- epilogue: residual add + store (f4_out || f6_out) ----------------
  {
    const int nn = lane & 15;
    #pragma unroll
    for(int r=0;r<8;++r){
      int M   = lo ? r : 8+r;
      int pix = g*16 + M;
      float v0 = oa[r] + pf[wave][M][nn];
      if (nn < 9) out[pix*9 + nn] = v0;
      else        out[NINE9 + pix*13 + (nn-9)] = v0;
      if (nn < 6){
        float v1 = ob[r] + pf[wave][M][16+nn];
        out[NINE9 + pix*13 + (16+nn-9)] = v1;
      }
    }
  }
}

extern "C" void kernel_launch(void* const* d_in, const int* in_sizes, int n_in,
                              void* d_out, int out_size, void* d_ws, size_t ws_size,
                              hipStream_t stream){
  const float* f4    = (const float*)d_in[0];
  const float* f6    = (const float*)d_in[1];
  const float* w_agg = (const float*)d_in[4];
  const float* w_tp  = (const float*)d_in[5];
  float* frg = (float*)d_ws;
  float* out = (float*)d_out;

  eq_setup<<<36, 256, 0, stream>>>(w_agg, w_tp, frg);
  eq_main<<<NPIX/128, 256, 0, stream>>>(f4, f6, frg, out);
  if (out_size >= NPIX*22 + 2)          // in case the (Hr, Wr) tuple is flattened too
    eq_tail<<<1, 1, 0, stream>>>(out + (size_t)NPIX*22);
}